// GNN_59468117180545
// MI455X (gfx1250) — compile-verified
//
#include <hip/hip_runtime.h>
#include <hip/hip_bf16.h>

#define N_NODES 50000
#define N_EDGES 800000
#define HIDDEN  128
#define HEADS   4
#define HCOL    (HEADS * HIDDEN)   // 512
#define LAYERS  2
#define NEG_SLOPE 0.2f

typedef float v2f __attribute__((ext_vector_type(2)));
typedef float v8f __attribute__((ext_vector_type(8)));

// ---- order-preserving float <-> uint encoding for atomicMax on floats ----
__device__ __forceinline__ unsigned enc_f(float f) {
    unsigned u = __float_as_uint(f);
    return (u & 0x80000000u) ? ~u : (u | 0x80000000u);
}
__device__ __forceinline__ float dec_f(unsigned u) {
    return (u & 0x80000000u) ? __uint_as_float(u ^ 0x80000000u)
                             : __uint_as_float(~u);
}

// =====================================================================
// GEMM1: H[N_NODES, 512] = X[N_NODES,128] @ W[128,512]
// One wave computes a 16x64 strip: 4 N-tiles reuse one A fragment.
// fp32 WMMA 16x16x4 (exact fp32 math).
// =====================================================================
__global__ void gemm_h_kernel(const float* __restrict__ X,
                              const float* __restrict__ W,
                              float* __restrict__ H) {
    const int wave = (blockIdx.x * blockDim.x + threadIdx.x) >> 5;
    const int lane = threadIdx.x & 31;
    const int MT   = N_NODES / 16;   // 3125
    const int NTG  = HCOL / 64;      // 8 groups of 4 tiles
    if (wave >= MT * NTG) return;
    const int mt = wave / NTG;
    const int ng = wave % NTG;
    const int l  = lane & 15;
    const int hi = lane >> 4;        // 0: lanes 0-15, 1: lanes 16-31
    const int m0 = mt * 16;
    const int n0 = ng * 64;

    v8f c0 = {}, c1 = {}, c2 = {}, c3 = {};
    const float* arow = X + (m0 + l) * HIDDEN;

    for (int k = 0; k < HIDDEN; k += 4) {
        v2f a;
        a[0] = arow[k + 2 * hi];
        a[1] = arow[k + 2 * hi + 1];
        const float* w0 = W + (k + 2 * hi) * HCOL + n0 + l;
        const float* w1 = W + (k + 2 * hi + 1) * HCOL + n0 + l;
        v2f b;
        b[0] = w0[0];  b[1] = w1[0];
        c0 = __builtin_amdgcn_wmma_f32_16x16x4_f32(false, a, false, b, (short)0, c0, false, false);
        b[0] = w0[16]; b[1] = w1[16];
        c1 = __builtin_amdgcn_wmma_f32_16x16x4_f32(false, a, false, b, (short)0, c1, false, false);
        b[0] = w0[32]; b[1] = w1[32];
        c2 = __builtin_amdgcn_wmma_f32_16x16x4_f32(false, a, false, b, (short)0, c2, false, false);
        b[0] = w0[48]; b[1] = w1[48];
        c3 = __builtin_amdgcn_wmma_f32_16x16x4_f32(false, a, false, b, (short)0, c3, false, false);
    }

    // D layout: VGPR j holds row M = (hi?8:0)+j, col N = n0(+tile*16) + l
    #pragma unroll
    for (int j = 0; j < 8; ++j) {
        float* drow = H + (m0 + hi * 8 + j) * HCOL + n0 + l;
        drow[0]  = c0[j];
        drow[16] = c1[j];
        drow[32] = c2[j];
        drow[48] = c3[j];
    }
}

// =====================================================================
// GEMM2: Out[N,128] = R@Pw + Xin@Sw + (Pb + Sb)
// Same structure, 2 groups of 4 N-tiles cover N=128.
// =====================================================================
__global__ void gemm_out_kernel(const float* __restrict__ R,
                                const float* __restrict__ Pw,
                                const float* __restrict__ Xin,
                                const float* __restrict__ Sw,
                                const float* __restrict__ Pb,
                                const float* __restrict__ Sb,
                                float* __restrict__ Out) {
    const int wave = (blockIdx.x * blockDim.x + threadIdx.x) >> 5;
    const int lane = threadIdx.x & 31;
    const int MT   = N_NODES / 16;   // 3125
    const int NTG  = HIDDEN / 64;    // 2
    if (wave >= MT * NTG) return;
    const int mt = wave / NTG;
    const int ng = wave % NTG;
    const int l  = lane & 15;
    const int hi = lane >> 4;
    const int m0 = mt * 16;
    const int n0 = ng * 64;

    v8f c0 = {}, c1 = {}, c2 = {}, c3 = {};

    const float* arow = R + (m0 + l) * HIDDEN;
    for (int k = 0; k < HIDDEN; k += 4) {
        v2f a;
        a[0] = arow[k + 2 * hi];
        a[1] = arow[k + 2 * hi + 1];
        const float* w0 = Pw + (k + 2 * hi) * HIDDEN + n0 + l;
        const float* w1 = Pw + (k + 2 * hi + 1) * HIDDEN + n0 + l;
        v2f b;
        b[0] = w0[0];  b[1] = w1[0];
        c0 = __builtin_amdgcn_wmma_f32_16x16x4_f32(false, a, false, b, (short)0, c0, false, false);
        b[0] = w0[16]; b[1] = w1[16];
        c1 = __builtin_amdgcn_wmma_f32_16x16x4_f32(false, a, false, b, (short)0, c1, false, false);
        b[0] = w0[32]; b[1] = w1[32];
        c2 = __builtin_amdgcn_wmma_f32_16x16x4_f32(false, a, false, b, (short)0, c2, false, false);
        b[0] = w0[48]; b[1] = w1[48];
        c3 = __builtin_amdgcn_wmma_f32_16x16x4_f32(false, a, false, b, (short)0, c3, false, false);
    }

    arow = Xin + (m0 + l) * HIDDEN;
    for (int k = 0; k < HIDDEN; k += 4) {
        v2f a;
        a[0] = arow[k + 2 * hi];
        a[1] = arow[k + 2 * hi + 1];
        const float* w0 = Sw + (k + 2 * hi) * HIDDEN + n0 + l;
        const float* w1 = Sw + (k + 2 * hi + 1) * HIDDEN + n0 + l;
        v2f b;
        b[0] = w0[0];  b[1] = w1[0];
        c0 = __builtin_amdgcn_wmma_f32_16x16x4_f32(false, a, false, b, (short)0, c0, false, false);
        b[0] = w0[16]; b[1] = w1[16];
        c1 = __builtin_amdgcn_wmma_f32_16x16x4_f32(false, a, false, b, (short)0, c1, false, false);
        b[0] = w0[32]; b[1] = w1[32];
        c2 = __builtin_amdgcn_wmma_f32_16x16x4_f32(false, a, false, b, (short)0, c2, false, false);
        b[0] = w0[48]; b[1] = w1[48];
        c3 = __builtin_amdgcn_wmma_f32_16x16x4_f32(false, a, false, b, (short)0, c3, false, false);
    }

    const float b0 = Pb[n0 + l]      + Sb[n0 + l];
    const float b1 = Pb[n0 + l + 16] + Sb[n0 + l + 16];
    const float b2 = Pb[n0 + l + 32] + Sb[n0 + l + 32];
    const float b3 = Pb[n0 + l + 48] + Sb[n0 + l + 48];
    #pragma unroll
    for (int j = 0; j < 8; ++j) {
        float* drow = Out + (m0 + hi * 8 + j) * HIDDEN + n0 + l;
        drow[0]  = c0[j] + b0;
        drow[16] = c1[j] + b1;
        drow[32] = c2[j] + b2;
        drow[48] = c3[j] + b3;
    }
}

// =====================================================================
// alpha_src[n,h] = dot(H[n,h,:], a_s[h,:]); alpha_dst likewise.
// One wave32 per (node, head); lanes each take a float4 slice.
// =====================================================================
__global__ void alpha_kernel(const float* __restrict__ H,
                             const float* __restrict__ as_,
                             const float* __restrict__ ad_,
                             float* __restrict__ asrc,
                             float* __restrict__ adst) {
    const int wave = (blockIdx.x * blockDim.x + threadIdx.x) >> 5;
    const int lane = threadIdx.x & 31;
    if (wave >= N_NODES * HEADS) return;
    const int n = wave / HEADS;
    const int h = wave % HEADS;
    const float4* hv = (const float4*)(H + (size_t)n * HCOL + h * HIDDEN);
    const float4* av = (const float4*)(as_ + h * HIDDEN);
    const float4* dv = (const float4*)(ad_ + h * HIDDEN);
    float4 x = hv[lane], a = av[lane], d = dv[lane];
    float s = x.x * a.x + x.y * a.y + x.z * a.z + x.w * a.w;
    float t = x.x * d.x + x.y * d.y + x.z * d.z + x.w * d.w;
    #pragma unroll
    for (int off = 16; off > 0; off >>= 1) {
        s += __shfl_xor(s, off, 32);
        t += __shfl_xor(t, off, 32);
    }
    if (lane == 0) {
        asrc[wave] = s;
        adst[wave] = t;
    }
}

// =====================================================================
// Per-(edge, head): alpha = leaky_relu(asrc[src]+adst[dst]); seg-max
// =====================================================================
__global__ void edge_max_kernel(const int* __restrict__ ei,
                                const float* __restrict__ asrc,
                                const float* __restrict__ adst,
                                float* __restrict__ ealpha,
                                unsigned* __restrict__ amax) {
    const int t = blockIdx.x * blockDim.x + threadIdx.x;
    if (t >= N_EDGES * HEADS) return;
    const int e = t >> 2;
    const int h = t & 3;
    const int s = ei[e];
    const int d = ei[N_EDGES + e];
    float a = asrc[s * HEADS + h] + adst[d * HEADS + h];
    a = (a >= 0.f) ? a : NEG_SLOPE * a;
    ealpha[t] = a;
    atomicMax(&amax[d * HEADS + h], enc_f(a));
}

// =====================================================================
// ex = exp(alpha - amax[dst]); denom[dst] += ex  (ex overwrites ealpha)
// =====================================================================
__global__ void edge_exp_kernel(const int* __restrict__ ei,
                                float* __restrict__ ealpha,
                                const unsigned* __restrict__ amax,
                                float* __restrict__ denom) {
    const int t = blockIdx.x * blockDim.x + threadIdx.x;
    if (t >= N_EDGES * HEADS) return;
    const int e = t >> 2;
    const int h = t & 3;
    const int d = ei[N_EDGES + e];
    const float ex = __expf(ealpha[t] - dec_f(amax[d * HEADS + h]));
    ealpha[t] = ex;
    atomicAdd(&denom[d * HEADS + h], ex);
}

// =====================================================================
// Aggregate: agg[dst,c] += (1/H) * sum_h w_{e,h} * H[src,h,c]
// One wave32 per edge; lane handles 4 channels; heads folded locally.
// Gather of H hits L2 (102 MB < 192 MB L2).
// =====================================================================
__global__ void edge_agg_kernel(const int* __restrict__ ei,
                                const float* __restrict__ ealpha,
                                const float* __restrict__ denom,
                                const float* __restrict__ H,
                                float* __restrict__ agg) {
    const int wave = (blockIdx.x * blockDim.x + threadIdx.x) >> 5;
    const int lane = threadIdx.x & 31;
    if (wave >= N_EDGES) return;
    const int e = wave;
    const int s = ei[e];
    const int d = ei[N_EDGES + e];

    float w[HEADS];
    #pragma unroll
    for (int h = 0; h < HEADS; ++h)
        w[h] = 0.25f * ealpha[e * HEADS + h] / (denom[d * HEADS + h] + 1e-16f);

    const float4* hv = (const float4*)(H + (size_t)s * HCOL);
    float4 acc = make_float4(0.f, 0.f, 0.f, 0.f);
    #pragma unroll
    for (int h = 0; h < HEADS; ++h) {
        float4 v = hv[h * (HIDDEN / 4) + lane];
        acc.x += w[h] * v.x;
        acc.y += w[h] * v.y;
        acc.z += w[h] * v.z;
        acc.w += w[h] * v.w;
    }
    float* o = agg + (size_t)d * HIDDEN + lane * 4;
    atomicAdd(o + 0, acc.x);
    atomicAdd(o + 1, acc.y);
    atomicAdd(o + 2, acc.z);
    atomicAdd(o + 3, acc.w);
}

// =====================================================================
// In-place: agg = relu(agg + conv_b)
// =====================================================================
__global__ void finalize_kernel(float* __restrict__ agg,
                                const float* __restrict__ bias) {
    const int t = blockIdx.x * blockDim.x + threadIdx.x;
    if (t >= N_NODES * HIDDEN) return;
    const float v = agg[t] + bias[t & (HIDDEN - 1)];
    agg[t] = v > 0.f ? v : 0.f;
}

// =====================================================================
extern "C" void kernel_launch(void* const* d_in, const int* in_sizes, int n_in,
                              void* d_out, int out_size, void* d_ws, size_t ws_size,
                              hipStream_t stream) {
    const float* x       = (const float*)d_in[0];
    const int*   ei      = (const int*)  d_in[1];
    const float* conv_w  = (const float*)d_in[2];
    const float* att_src = (const float*)d_in[3];
    const float* att_dst = (const float*)d_in[4];
    const float* conv_b  = (const float*)d_in[5];
    const float* proj_w  = (const float*)d_in[6];
    const float* proj_b  = (const float*)d_in[7];
    const float* skip_w  = (const float*)d_in[8];
    const float* skip_b  = (const float*)d_in[9];
    float* out = (float*)d_out;

    // ---- workspace carve-up ----
    char* ws = (char*)d_ws;
    float*    H      = (float*)ws;                ws += (size_t)N_NODES * HCOL  * 4;  // 102.4 MB
    float*    xbuf   = (float*)ws;                ws += (size_t)N_NODES * HIDDEN * 4; //  25.6 MB
    float*    agg    = (float*)ws;                ws += (size_t)N_NODES * HIDDEN * 4; //  25.6 MB
    float*    ealpha = (float*)ws;                ws += (size_t)N_EDGES * HEADS * 4;  //  12.8 MB
    float*    asrc   = (float*)ws;                ws += (size_t)N_NODES * HEADS * 4;
    float*    adst   = (float*)ws;                ws += (size_t)N_NODES * HEADS * 4;
    unsigned* amax   = (unsigned*)ws;             ws += (size_t)N_NODES * HEADS * 4;
    float*    denom  = (float*)ws;                ws += (size_t)N_NODES * HEADS * 4;

    for (int l = 0; l < LAYERS; ++l) {
        const float* xin  = (l == 0) ? x : xbuf;
        float*       xout = (l == LAYERS - 1) ? out : xbuf;
        const float* Wl  = conv_w  + (size_t)l * HIDDEN * HCOL;
        const float* asl = att_src + (size_t)l * HEADS * HIDDEN;
        const float* adl = att_dst + (size_t)l * HEADS * HIDDEN;
        const float* cbl = conv_b  + (size_t)l * HIDDEN;
        const float* pwl = proj_w  + (size_t)l * HIDDEN * HIDDEN;
        const float* pbl = proj_b  + (size_t)l * HIDDEN;
        const float* swl = skip_w  + (size_t)l * HIDDEN * HIDDEN;
        const float* sbl = skip_b  + (size_t)l * HIDDEN;

        // 1) H = xin @ W   (16x64 strip per wave; 25000 waves, 4 waves/block)
        gemm_h_kernel<<<6250, 128, 0, stream>>>(xin, Wl, H);

        // 2) per-node attention logits
        alpha_kernel<<<(N_NODES * HEADS + 7) / 8, 256, 0, stream>>>(H, asl, adl, asrc, adst);

        // 3) zero reduction buffers (graph-capturable memset nodes)
        hipMemsetAsync(amax,  0, (size_t)N_NODES * HEADS * 4, stream);
        hipMemsetAsync(denom, 0, (size_t)N_NODES * HEADS * 4, stream);
        hipMemsetAsync(agg,   0, (size_t)N_NODES * HIDDEN * 4, stream);

        // 4) segment max / 5) exp + segment sum
        edge_max_kernel<<<(N_EDGES * HEADS + 255) / 256, 256, 0, stream>>>(ei, asrc, adst, ealpha, amax);
        edge_exp_kernel<<<(N_EDGES * HEADS + 255) / 256, 256, 0, stream>>>(ei, ealpha, amax, denom);

        // 6) weighted scatter-add aggregation (wave per edge)
        edge_agg_kernel<<<N_EDGES / 8, 256, 0, stream>>>(ei, ealpha, denom, H, agg);

        // 7) bias + relu in place
        finalize_kernel<<<(N_NODES * HIDDEN + 255) / 256, 256, 0, stream>>>(agg, cbl);

        // 8) xout = relu_conv @ proj_w + xin @ skip_w + (proj_b + skip_b)
        gemm_out_kernel<<<1563, 128, 0, stream>>>(agg, pwl, xin, swl, pbl, sbl, xout);
    }
}